// LlamaDecoderLayer_59115929862736
// MI455X (gfx1250) — compile-verified
//
#include <hip/hip_runtime.h>
#include <math.h>

// Llama decoder layer, B=16 decode step, fp32, MI455X (gfx1250, wave32).
// Memory-bound (~1.25 GB/call @ 23.3 TB/s -> ~54us floor). All projections use
// V_WMMA_F32_16X16X4_F32 with M=16 (= batch), split-K + fp32 global atomics.
// Weight streams are single-use -> non-temporal loads + explicit prefetch.

#define H      4096
#define NHEADS 32
#define NKV    8
#define HD     128
#define GRP    4
#define FFN    11008
#define BB     16
#define SC     4096
#define STOT   4097          // cache + new token
#define SP     4112          // padded scores row stride (multiple of 16)
#define EPS_   1e-5f
#define QKVW   (H + 2*NKV*HD)   // 6144 fused q|k_new|v_new row width

typedef __attribute__((ext_vector_type(2))) float v2f;
typedef __attribute__((ext_vector_type(4))) float v4f;
typedef __attribute__((ext_vector_type(8))) float v8f;

// ---------------- utility kernels ----------------

__global__ __launch_bounds__(256) void zero_kernel(float* __restrict__ p, int n) {
    int i = blockIdx.x * 256 + threadIdx.x;
    if (i < n) p[i] = 0.f;
}

__global__ __launch_bounds__(256) void copy_kernel(float* __restrict__ dst,
                                                   const float* __restrict__ src, int n) {
    int i = blockIdx.x * 256 + threadIdx.x;
    if (i < n) dst[i] = src[i];
}

// ---------------- RMSNorm: y[b,:] = w * x[b,:] * rsqrt(mean(x^2)+eps) ----------------

__global__ __launch_bounds__(256) void rms_kernel(const float* __restrict__ x,
                                                  const float* __restrict__ w,
                                                  float* __restrict__ y) {
    int b = blockIdx.x, tid = threadIdx.x;
    int wid = tid >> 5, lane = tid & 31;
    const float* xr = x + (size_t)b * H;
    float s = 0.f;
    for (int i = tid; i < H; i += 256) { float v = xr[i]; s += v * v; }
    for (int off = 16; off; off >>= 1) s += __shfl_xor(s, off, 32);
    __shared__ float red[8];
    if (lane == 0) red[wid] = s;
    __syncthreads();
    if (tid == 0) {
        float t = 0.f;
        for (int i = 0; i < 8; i++) t += red[i];
        red[0] = rsqrtf(t / (float)H + EPS_);
    }
    __syncthreads();
    float rn = red[0];
    for (int i = tid; i < H; i += 256) y[(size_t)b * H + i] = w[i] * xr[i] * rn;
}

// ---------------- WMMA GEMM: y[16,N] += x[16,K] @ W[N,K]^T  (split-K, atomic) ----------------
// A frag (16x4 f32): lane l holds row  m=(l&15), K pair at k0+2*(l>>4)  -> v2f load
// B frag (4x16 f32): lane l holds col  n=(l&15), K pair at k0+2*(l>>4)  -> v2f load
// C (16x16 f32, 8 VGPRs): element r -> row r+8*(l>>4), col n0+(l&15)

#define WMMA4(A, Bv, C) __builtin_amdgcn_wmma_f32_16x16x4_f32( \
        false, (A), false, (Bv), (short)0, (C), false, false)

__global__ __launch_bounds__(128) void gemm_wmma(const float* __restrict__ x,
                                                 const float* __restrict__ W,
                                                 float* __restrict__ y,
                                                 int K, int N, int ldo, int kchunk) {
    int wid = threadIdx.x >> 5, lane = threadIdx.x & 31;
    int n0 = (blockIdx.x * 4 + wid) * 16;
    if (n0 >= N) return;                       // wave-uniform exit
    int kbeg = blockIdx.y * kchunk;
    int kend = min(K, kbeg + kchunk);
    int hl = lane >> 4, ll = lane & 15;

    const float* pa = x + (size_t)ll * K + hl * 2 + kbeg;
    const float* pb = W + (size_t)(n0 + ll) * K + hl * 2 + kbeg;

    int nk = kend - kbeg;
    int k16 = nk & ~15;

    v8f acc = {0.f, 0.f, 0.f, 0.f, 0.f, 0.f, 0.f, 0.f};
    for (int k = 0; k < k16; k += 16) {
        // 8 clustered b64 loads (activations cached, weights non-temporal)
        v2f a0 = *reinterpret_cast<const v2f*>(pa + k);
        v2f a1 = *reinterpret_cast<const v2f*>(pa + k + 4);
        v2f a2 = *reinterpret_cast<const v2f*>(pa + k + 8);
        v2f a3 = *reinterpret_cast<const v2f*>(pa + k + 12);
        v2f b0 = __builtin_nontemporal_load(reinterpret_cast<const v2f*>(pb + k));
        v2f b1 = __builtin_nontemporal_load(reinterpret_cast<const v2f*>(pb + k + 4));
        v2f b2 = __builtin_nontemporal_load(reinterpret_cast<const v2f*>(pb + k + 8));
        v2f b3 = __builtin_nontemporal_load(reinterpret_cast<const v2f*>(pb + k + 12));
        __builtin_prefetch(pb + k + 1024, 0, 1);   // weight stream ~4KB ahead
        acc = WMMA4(a0, b0, acc);
        acc = WMMA4(a1, b1, acc);
        acc = WMMA4(a2, b2, acc);
        acc = WMMA4(a3, b3, acc);
    }
    for (int k = k16; k < nk; k += 4) {            // ragged tail
        v2f a = *reinterpret_cast<const v2f*>(pa + k);
        v2f b = *reinterpret_cast<const v2f*>(pb + k);
        acc = WMMA4(a, b, acc);
    }

    int col = n0 + ll;
#pragma unroll
    for (int r = 0; r < 8; r++) {
        int row = r + 8 * hl;
        atomicAdd(&y[(size_t)row * ldo + col], acc[r]);
    }
}

// ---------------- attention: scores[b,kv,g,s] = (q . k_s) * scale ----------------

__global__ __launch_bounds__(256) void scores_kernel(const float* __restrict__ k_cache,
                                                     const float* __restrict__ qkv,
                                                     float* __restrict__ scores) {
    int b = blockIdx.x, kv = blockIdx.y;
    int wid = threadIdx.x >> 5, lane = threadIdx.x & 31;
    const float scale = 0.08838834764831845f;   // 1/sqrt(128)

    v4f qv[GRP];
    const float* qb = qkv + (size_t)b * QKVW + (kv * GRP) * HD + lane * 4;
#pragma unroll
    for (int g = 0; g < GRP; g++) qv[g] = *reinterpret_cast<const v4f*>(qb + g * HD);

    int sbase = blockIdx.z * 256 + wid * 32;
    for (int i = 0; i < 32; i++) {
        int s = sbase + i;
        if (s >= STOT) break;
        const float* krow = (s < SC)
            ? (k_cache + ((size_t)(b * NKV + kv) * SC + s) * HD)
            : (qkv + (size_t)b * QKVW + H + kv * HD);          // new k token
        v4f k4 = __builtin_nontemporal_load(
            reinterpret_cast<const v4f*>(krow + lane * 4));    // K row single-use
        float p[GRP];
#pragma unroll
        for (int g = 0; g < GRP; g++)
            p[g] = qv[g].x * k4.x + qv[g].y * k4.y + qv[g].z * k4.z + qv[g].w * k4.w;
#pragma unroll
        for (int g = 0; g < GRP; g++)
            for (int off = 16; off; off >>= 1) p[g] += __shfl_xor(p[g], off, 32);
        if (lane == 0) {
            size_t base = (size_t)((b * NKV + kv) * GRP) * SP + s;
#pragma unroll
            for (int g = 0; g < GRP; g++) scores[base + (size_t)g * SP] = p[g] * scale;
        }
    }
}

// ---------------- softmax over one (b,kv,g) row, in place ----------------

__global__ __launch_bounds__(256) void softmax_kernel(float* __restrict__ scores) {
    int tid = threadIdx.x, wid = tid >> 5, lane = tid & 31;
    float* p = scores + (size_t)blockIdx.x * SP;
    __shared__ float red[8];

    float m = -1e30f;
    for (int i = tid; i < STOT; i += 256) m = fmaxf(m, p[i]);
    for (int off = 16; off; off >>= 1) m = fmaxf(m, __shfl_xor(m, off, 32));
    if (lane == 0) red[wid] = m;
    __syncthreads();
    if (tid == 0) { float t = red[0]; for (int i = 1; i < 8; i++) t = fmaxf(t, red[i]); red[0] = t; }
    __syncthreads();
    m = red[0];
    __syncthreads();

    float s = 0.f;
    for (int i = tid; i < STOT; i += 256) { float e = expf(p[i] - m); p[i] = e; s += e; }
    for (int off = 16; off; off >>= 1) s += __shfl_xor(s, off, 32);
    if (lane == 0) red[wid] = s;
    __syncthreads();
    if (tid == 0) { float t = 0.f; for (int i = 0; i < 8; i++) t += red[i]; red[0] = t; }
    __syncthreads();
    float inv = 1.f / red[0];
    for (int i = tid; i < STOT; i += 256) p[i] *= inv;
}

// ---------------- attn[b,kv,g,d] += sum_s probs * v  (split-S, atomic) ----------------

__global__ __launch_bounds__(128) void pv_kernel(const float* __restrict__ v_cache,
                                                 const float* __restrict__ qkv,
                                                 const float* __restrict__ probs,
                                                 float* __restrict__ attn) {
    int b = blockIdx.x, kv = blockIdx.y, d = threadIdx.x;
    int sbeg = blockIdx.z * 512;
    int send = min(STOT, sbeg + 512);
    float acc[GRP] = {0.f, 0.f, 0.f, 0.f};
    size_t pbase = (size_t)((b * NKV + kv) * GRP) * SP;
    for (int s = sbeg; s < send; s++) {
        const float* vrow = (s < SC)
            ? (v_cache + ((size_t)(b * NKV + kv) * SC + s) * HD)
            : (qkv + (size_t)b * QKVW + H + NKV * HD + kv * HD);   // new v token
        float vv = __builtin_nontemporal_load(vrow + d);           // V row single-use
#pragma unroll
        for (int g = 0; g < GRP; g++) acc[g] += probs[pbase + (size_t)g * SP + s] * vv;
    }
#pragma unroll
    for (int g = 0; g < GRP; g++)
        atomicAdd(&attn[(size_t)b * H + (kv * GRP + g) * HD + d], acc[g]);
}

// ---------------- SwiGLU: m = silu(g) * u ----------------

__global__ __launch_bounds__(256) void silu_kernel(const float* __restrict__ g,
                                                   const float* __restrict__ u,
                                                   float* __restrict__ m, int n) {
    int i = blockIdx.x * 256 + threadIdx.x;
    if (i < n) { float gv = g[i]; m[i] = (gv / (1.f + expf(-gv))) * u[i]; }
}

// ---------------- launcher ----------------

extern "C" void kernel_launch(void* const* d_in, const int* in_sizes, int n_in,
                              void* d_out, int out_size, void* d_ws, size_t ws_size,
                              hipStream_t stream) {
    const float* hidden  = (const float*)d_in[0];
    const float* k_cache = (const float*)d_in[1];
    const float* v_cache = (const float*)d_in[2];
    const float* Wq = (const float*)d_in[3];
    const float* Wk = (const float*)d_in[4];
    const float* Wv = (const float*)d_in[5];
    const float* Wo = (const float*)d_in[6];
    const float* Wg = (const float*)d_in[7];
    const float* Wu = (const float*)d_in[8];
    const float* Wd = (const float*)d_in[9];
    const float* ln1 = (const float*)d_in[10];
    const float* ln2 = (const float*)d_in[11];
    float* out = (float*)d_out;

    float* ws     = (float*)d_ws;
    float* h1     = ws;                  // 16*4096
    float* qkv    = h1   + BB * H;       // 16*6144
    float* attn   = qkv  + BB * QKVW;    // 16*4096
    float* h2     = attn + BB * H;       // 16*4096
    float* h3     = h2   + BB * H;       // 16*4096
    float* gbuf   = h3   + BB * H;       // 16*11008
    float* ubuf   = gbuf + BB * FFN;     // 16*11008
    float* mbuf   = ubuf + BB * FFN;     // 16*11008
    float* scores = mbuf + BB * FFN;     // 512*4112  (~12 MB total ws use)

    auto gemm = [&](const float* x, const float* W, float* y,
                    int K, int N, int ldo, int ksplit) {
        int kchunk = (((K + ksplit - 1) / ksplit) + 15) & ~15;
        dim3 grid((N / 16 + 3) / 4, ksplit);
        gemm_wmma<<<grid, dim3(128), 0, stream>>>(x, W, y, K, N, ldo, kchunk);
    };

    // ---- attention block ----
    rms_kernel<<<BB, 256, 0, stream>>>(hidden, ln1, h1);
    zero_kernel<<<(BB * QKVW + 255) / 256, 256, 0, stream>>>(qkv, BB * QKVW);
    gemm(h1, Wq, qkv,              H, H,        QKVW, 8);
    gemm(h1, Wk, qkv + H,          H, NKV * HD, QKVW, 8);
    gemm(h1, Wv, qkv + H + NKV*HD, H, NKV * HD, QKVW, 8);

    scores_kernel<<<dim3(BB, NKV, (STOT + 255) / 256), 256, 0, stream>>>(k_cache, qkv, scores);
    softmax_kernel<<<BB * NKV * GRP, 256, 0, stream>>>(scores);
    zero_kernel<<<(BB * H + 255) / 256, 256, 0, stream>>>(attn, BB * H);
    pv_kernel<<<dim3(BB, NKV, (STOT + 511) / 512), 128, 0, stream>>>(v_cache, qkv, scores, attn);

    copy_kernel<<<(BB * H + 255) / 256, 256, 0, stream>>>(h2, hidden, BB * H);
    gemm(attn, Wo, h2, H, H, H, 8);                      // residual + o_proj

    // ---- MLP block ----
    rms_kernel<<<BB, 256, 0, stream>>>(h2, ln2, h3);
    zero_kernel<<<(2 * BB * FFN + 255) / 256, 256, 0, stream>>>(gbuf, 2 * BB * FFN);
    gemm(h3, Wg, gbuf, H, FFN, FFN, 8);
    gemm(h3, Wu, ubuf, H, FFN, FFN, 8);
    silu_kernel<<<(BB * FFN + 255) / 256, 256, 0, stream>>>(gbuf, ubuf, mbuf, BB * FFN);

    copy_kernel<<<(BB * H + 255) / 256, 256, 0, stream>>>(out, h2, BB * H);
    gemm(mbuf, Wd, out, FFN, H, H, 16);                  // residual + down_proj
}